// PeakAligner_62053687492851
// MI455X (gfx1250) — compile-verified
//
#include <hip/hip_runtime.h>
#include <stdint.h>

// ---- reference constants ----
#define L_IN        220   // row length
#define IPS         64    // output samples per row
#define FACTOR      3
#define LO_W        57    // window lo == NEW_CENTER
#define WIN         30    // window length [57,87)
#define ALIGNED_LEN 192   // IPS*FACTOR
#define MAX_START   28    // L_IN - ALIGNED_LEN
#define NEW_CENTER  57

#define ROWS_PER_BLOCK 8
#define LDS_STRIDE_F   224   // 220 floats padded to 16B multiple (896 B / wave)

__global__ __launch_bounds__(256)
void PeakAligner_62053687492851_kernel(const float* __restrict__ in,
                                       float* __restrict__ out,
                                       float* __restrict__ maskout,
                                       int K)
{
    __shared__ float smem[ROWS_PER_BLOCK * LDS_STRIDE_F];

    const int lane = threadIdx.x & 31;
    const int w    = threadIdx.x >> 5;           // wave id in block
    const int row  = blockIdx.x * ROWS_PER_BLOCK + w;
    if (row >= K) return;                        // wave-uniform; no block barrier used

    // ---------------- async global -> LDS staging of one 880B row ----------------
    // 55 chunks of 16B; lanes cover chunks {lane, min(lane+32,54)} (clamped so
    // EXEC stays all-ones and the duplicate writes of chunk 54 are benign).
    // Stream is read-once (440MB >> 192MB L2) -> non-temporal load hint.
    {
        const char* rowp   = (const char*)(in + (size_t)row * L_IN);
        unsigned ldsbase   = (unsigned)(uintptr_t)(&smem[w * LDS_STRIDE_F]);
        int c0 = lane;
        int c1 = (lane + 32 < 55) ? (lane + 32) : 54;
        uint64_t g0 = (uint64_t)(uintptr_t)rowp + (unsigned)(c0 * 16);
        uint64_t g1 = (uint64_t)(uintptr_t)rowp + (unsigned)(c1 * 16);
        unsigned l0 = ldsbase + (unsigned)(c0 * 16);
        unsigned l1 = ldsbase + (unsigned)(c1 * 16);
        asm volatile("global_load_async_to_lds_b128 %0, %1, off th:TH_LOAD_NT"
                     :: "v"(l0), "v"(g0) : "memory");
        asm volatile("global_load_async_to_lds_b128 %0, %1, off th:TH_LOAD_NT"
                     :: "v"(l1), "v"(g1) : "memory");
        asm volatile("s_wait_asynccnt 0x0" ::: "memory");
    }

    const float* rb = &smem[w * LDS_STRIDE_F];

    // ---------------- phase A: argmax over window [57,87) --------------------
    float v; int idx;
    if (lane < WIN) { v = rb[LO_W + lane]; idx = lane; }
    else            { v = -__builtin_inff(); idx = 0x7fffffff; }
    #pragma unroll
    for (int off = 16; off; off >>= 1) {
        float ov = __shfl_xor(v,   off, 32);
        int   oi = __shfl_xor(idx, off, 32);
        if (ov > v || (ov == v && oi < idx)) { v = ov; idx = oi; }
    }
    const int  start      = idx;                   // index_max - 57, in [0,29]
    const bool in_bounds  = (start <= MAX_START);
    const int  safe_start = (start > MAX_START) ? MAX_START : start;

    // ---------------- phase B: argmax over aligned[0:192) --------------------
    float bv = rb[safe_start + lane]; int bidx = lane;
    #pragma unroll
    for (int i = 1; i < 6; ++i) {                  // 192 = 32*6, indices ascending
        int j = lane + 32 * i;
        float x = rb[safe_start + j];
        if (x > bv) { bv = x; bidx = j; }          // strict > keeps first occurrence
    }
    #pragma unroll
    for (int off = 16; off; off >>= 1) {
        float ov = __shfl_xor(bv,   off, 32);
        int   oi = __shfl_xor(bidx, off, 32);
        if (ov > bv || (ov == bv && oi < bidx)) { bv = ov; bidx = oi; }
    }
    const bool valid = in_bounds && (bidx == NEW_CENTER);

    // ---------------- phase C: downsample + mask (write-once -> NT stores) ---
    float* orow = out + (size_t)row * IPS;
    const int j0 = lane, j1 = lane + 32;           // stride-3 dword reads: no bank conflicts
    __builtin_nontemporal_store(valid ? rb[safe_start + FACTOR * j0] : 0.0f, &orow[j0]);
    __builtin_nontemporal_store(valid ? rb[safe_start + FACTOR * j1] : 0.0f, &orow[j1]);
    if (lane == 0)
        __builtin_nontemporal_store(valid ? 0.0f : 1.0f, &maskout[row]); // removed_mask = !valid
}

extern "C" void kernel_launch(void* const* d_in, const int* in_sizes, int n_in,
                              void* d_out, int out_size, void* d_ws, size_t ws_size,
                              hipStream_t stream)
{
    const float* in = (const float*)d_in[0];
    const int K = in_sizes[0] / L_IN;              // 500000
    float* out     = (float*)d_out;                // K*64 floats (spikes_final)
    float* maskout = out + (size_t)K * IPS;        // then K mask values (0/1)
    const int blocks = (K + ROWS_PER_BLOCK - 1) / ROWS_PER_BLOCK;
    PeakAligner_62053687492851_kernel<<<blocks, 256, 0, stream>>>(in, out, maskout, K);
}